// Sine_Generator_Fast_52381421142259
// MI455X (gfx1250) — compile-verified
//
#include <hip/hip_runtime.h>

// Reference problem constants (match setup_inputs in the reference):
//   B=4, NHAR=64, T_FRAMES=256, HOP=512, SR=44100, win = 2*HOP = 1024
// hop_size / sampling_rate arrive as device scalars (d_in[3], d_in[4]).

typedef __attribute__((ext_vector_type(2))) float v2f;
typedef __attribute__((ext_vector_type(8))) float v8f;

#define PI_F 3.14159265358979323846f

// gfx1250 async global->LDS staging (ASYNCcnt path), guarded so the file
// compiles on toolchains that lack the builtin. Parameter types per the
// probe: (AS1 int*, AS3 int*, constant int offset, constant int cpol).
#if defined(__gfx1250__) && __has_builtin(__builtin_amdgcn_global_load_async_to_lds_b32)
#define HAVE_ASYNC_LDS 1
typedef __attribute__((address_space(1))) int* gas_i32p;
typedef __attribute__((address_space(3))) int* las_i32p;
#else
#define HAVE_ASYNC_LDS 0
#endif

__global__ void zero_f32_kernel(float* __restrict__ p, int n) {
    int i = blockIdx.x * blockDim.x + threadIdx.x;
    if (i < n) p[i] = 0.0f;
}

// One workgroup per frame (b,t). 128 threads = 4 wave32; wave w computes the
// 256-sample tile x in [256w, 256w+256) as a 16x16 matrix D[m][n],
// x = 256w + 16m + n, via two fp32 WMMA accumulations per K-chunk of 4
// harmonics (angle-addition factorization):
//   D = A1*B1 + A2*B2 over h = 0..63
//   A1[m,h] = cos(om_h*g_m + ph_h)   A2[m,h] = sin(om_h*g_m + ph_h)
//   B1[h,n] = a_h*cos(om_h*n)        B2[h,n] = -a_h*sin(om_h*n)
// B1/B2 are wave-invariant -> precomputed once per block into LDS.
__global__ void __launch_bounds__(128)
sine_olam_wmma_kernel(const float* __restrict__ ampl,
                      const float* __restrict__ phase,
                      const float* __restrict__ f0,
                      const int* __restrict__ hopp,
                      const int* __restrict__ srp,
                      float* __restrict__ out) {
    const int frame = blockIdx.x;     // b*256 + t
    const int t = frame & 255;
    const int b = frame >> 8;
    const int tid = threadIdx.x;
    const int w = tid >> 5;           // wave id 0..3 -> tile base 256*w
    const int L = tid & 31;           // lane id

    __shared__ float s_f0[64];        // raw f0 (scaled to omega at use)
    __shared__ float s_ph[64];
    __shared__ float s_am[64];
    // B fragments, [n][h] with row stride 68 floats -> conflict-free b64
    // loads in the K-loop (lanes hit banks 4*(L&15)+hb, all distinct).
    __shared__ alignas(16) float s_b1[16 * 68];
    __shared__ alignas(16) float s_b2[16 * 68];

    const int hop = hopp[0];          // 512
    const int sr  = srp[0];           // 44100
    const float omscale = 2.0f * PI_F / (float)sr;

    // ---- Stage per-frame parameters into LDS --------------------------
    if (tid < 64) {
        const int h = tid;
        const int gidx = (b * 64 + h) * 256 + t;   // [B,H,T] layout
#if HAVE_ASYNC_LDS
        __builtin_amdgcn_global_load_async_to_lds_b32(
            (gas_i32p)(ampl + gidx), (las_i32p)(s_am + h), 0, 0);
        __builtin_amdgcn_global_load_async_to_lds_b32(
            (gas_i32p)(phase + gidx), (las_i32p)(s_ph + h), 0, 0);
        __builtin_amdgcn_global_load_async_to_lds_b32(
            (gas_i32p)(f0 + gidx), (las_i32p)(s_f0 + h), 0, 0);
#else
        s_am[h] = ampl[gidx];
        s_ph[h] = phase[gidx];
        s_f0[h] = f0[gidx];
#endif
    }
#if HAVE_ASYNC_LDS
#if __has_builtin(__builtin_amdgcn_s_wait_asynccnt)
    __builtin_amdgcn_s_wait_asynccnt(0);
#else
    asm volatile("s_wait_asynccnt 0x0" ::: "memory");
#endif
#endif
    __syncthreads();

    // ---- Precompute wave-invariant B fragments (1024 unique sincos) ---
    for (int i = tid; i < 16 * 64; i += 128) {
        const int n = i >> 6;
        const int h = i & 63;
        float sn, cs;
        __sincosf(s_f0[h] * omscale * (float)n, &sn, &cs);
        const float a = s_am[h];
        s_b1[n * 68 + h] = a * cs;
        s_b2[n * 68 + h] = -a * sn;
    }
    __syncthreads();

    // Per-lane fragment roles (ISA 7.12.2, wave32):
    //  A 16x4 f32: lane L holds A[M=L&15, K=2*(L>>4)+{0,1}] in regs {x,y}
    //  B 4x16 f32: lane L holds B[K=2*(L>>4)+{0,1}, N=L&15] in regs {x,y}
    //  -> both fragments of lane L use harmonics h = 4*kc + 2*(L>>4) + {0,1}
    const int   m  = L & 15;
    const int   kh = (L >> 4) * 2;
    const float g  = (float)(256 * w + 16 * m - hop); // coarse pos (A role)
    const int   nb = m * 68;                          // B-frag LDS row base

    v8f acc = {0.f, 0.f, 0.f, 0.f, 0.f, 0.f, 0.f, 0.f};

    #pragma unroll
    for (int kc = 0; kc < 16; ++kc) {
        const int hb = kc * 4 + kh;
        const float om0 = s_f0[hb]     * omscale;
        const float om1 = s_f0[hb + 1] * omscale;
        const float ph0 = s_ph[hb],   ph1 = s_ph[hb + 1];

        v2f a1, a2;
        float sn, cs;
        __sincosf(om0 * g + ph0, &sn, &cs);  a1.x = cs;  a2.x = sn;
        __sincosf(om1 * g + ph1, &sn, &cs);  a1.y = cs;  a2.y = sn;

        const v2f b1 = *(const v2f*)(s_b1 + nb + hb);
        const v2f b2 = *(const v2f*)(s_b2 + nb + hb);

        // D += A2*B2 ; D += A1*B1   (fp32 16x16x4 WMMA, K=4 harmonics each)
        acc = __builtin_amdgcn_wmma_f32_16x16x4_f32(
                  false, a2, false, b2, (short)0, acc, false, false);
        acc = __builtin_amdgcn_wmma_f32_16x16x4_f32(
                  false, a1, false, b1, (short)0, acc, false, false);
    }

    // ---- Epilogue: Hann window + overlap-add --------------------------
    // Frame t contributes to out[t*hop + x - hop], x in [0, 2*hop).
    // Each output sample receives exactly <=2 contributions; two-operand
    // fp32 add is commutative, so atomic order never changes result bits.
    const int   win    = 2 * hop;
    const float wscale = 2.0f * PI_F / (float)win;
    const int   rowoff = (L >> 4) * 8;
    float* __restrict__ outb = out + b * (256 * hop);

    #pragma unroll
    for (int r = 0; r < 8; ++r) {
        const int M = r + rowoff;                 // C/D row for reg r
        const int x = 256 * w + 16 * M + (L & 15);
        const float hann = 0.5f - 0.5f * __cosf(wscale * (float)x);
        const int idx = t * hop + x - hop;
        if (idx >= 0) {
            atomicAdd(&outb[idx], acc[r] * hann);
        }
    }
}

extern "C" void kernel_launch(void* const* d_in, const int* in_sizes, int n_in,
                              void* d_out, int out_size, void* d_ws, size_t ws_size,
                              hipStream_t stream) {
    const float* ampl  = (const float*)d_in[0];
    const float* phase = (const float*)d_in[1];
    const float* f0    = (const float*)d_in[2];
    const int*   hopp  = (const int*)d_in[3];
    const int*   srp   = (const int*)d_in[4];
    float* out = (float*)d_out;

    // Zero the output every call (harness poisons once, we accumulate).
    {
        int threads = 256;
        int blocks = (out_size + threads - 1) / threads;
        zero_f32_kernel<<<blocks, threads, 0, stream>>>(out, out_size);
    }

    // B=4, T=256 -> 1024 frames, one block (4 wave32) per frame.
    const int B = 4, T = 256;
    sine_olam_wmma_kernel<<<B * T, 128, 0, stream>>>(ampl, phase, f0,
                                                     hopp, srp, out);
}